// GCNModel_30648886624547
// MI455X (gfx1250) — compile-verified
//
#include <hip/hip_runtime.h>
#include <hip/hip_bf16.h>

typedef float v2f __attribute__((ext_vector_type(2)));
typedef float v8f __attribute__((ext_vector_type(8)));

static constexpr int IN_DIM = 128;
static constexpr int HID    = 512;
static constexpr int H2     = 256;
static constexpr int H4     = 128;
static constexpr int NG     = 256;   // number of graphs

// ---------------------------------------------------------------------------
// fp32 WMMA GEMM with LDS staging: C[M,N] = act( A[M,K] @ B[K,N] + bias )
// Block = 256 threads = 8 waves. Block tile: 128(M) x 64(N), K chunked by 64.
// Each wave owns a 16(M) x 64(N) strip via V_WMMA_F32_16X16X4_F32.
// Requirements: N % 64 == 0, K % 64 == 0 (true for all layers here).
// ---------------------------------------------------------------------------
#define KC      64
#define ASTRIDE 68    // 64 + 4 pad: b64 fragment reads conflict-free, b128-store aligned
#define BSTRIDE 66    // 64 + 2 pad: transposed B, conflict-free b64 fragment reads

__global__ __launch_bounds__(256)
void gemm_wmma_f32(const float* __restrict__ A, const float* __restrict__ B,
                   const float* __restrict__ bias, float* __restrict__ C,
                   int M, int K, int N, int reluIn, int reluOut)
{
    __shared__ float Alds[128 * ASTRIDE];   // 34.0 KB
    __shared__ float Blds[64 * BSTRIDE];    // 16.5 KB

    const int t     = threadIdx.x;
    const int lane  = t & 31;
    const int wave  = t >> 5;
    const int mBase = blockIdx.x * 128;
    const int nTile = blockIdx.y * 64;

    const int mLocal = wave * 16 + (lane & 15);  // fragment row within block tile
    const int kHalf  = (lane >> 4) * 2;          // 0 (lanes 0-15) or 2 (16-31)
    const int nCol   = lane & 15;

    // ---- precompute all k0-invariant addressing (no 64-bit muls in loop) ----
    long aOff[8];  int aLdsOff[8];
#pragma unroll
    for (int i = 0; i < 8; ++i) {
        const int idx  = t + i * 256;          // 0..2047
        const int row  = idx >> 4;             // 0..127
        const int cq   = idx & 15;             // float4 index within row
        const int gRow = mBase + row;
        const int gR   = (gRow < M) ? gRow : (M - 1);
        aOff[i]    = (long)gR * K + (cq << 2);
        aLdsOff[i] = row * ASTRIDE + (cq << 2);
    }
    long bOff[4];  int bLdsOff[4];
#pragma unroll
    for (int i = 0; i < 4; ++i) {
        const int idx = t + i * 256;           // 0..1023
        const int r   = idx >> 4;              // k within chunk, 0..63
        const int cq  = idx & 15;              // float4 index within row
        bOff[i]    = (long)r * N + nTile + (cq << 2);
        bLdsOff[i] = (cq << 2) * BSTRIDE + r;  // transposed store base
    }

    v8f acc[4] = {v8f{}, v8f{}, v8f{}, v8f{}};
    const float* Bk = B;                       // advances by KC*N per chunk

    for (int k0 = 0; k0 < K; k0 += KC) {
        __syncthreads();   // previous chunk's LDS reads done

        // ---- stage A tile: 128 x 64, coalesced float4 loads, float4 stores ----
#pragma unroll
        for (int i = 0; i < 8; ++i) {
            float4 v = *(const float4*)(A + aOff[i] + k0);
            if (reluIn) {
                v.x = fmaxf(v.x, 0.f); v.y = fmaxf(v.y, 0.f);
                v.z = fmaxf(v.z, 0.f); v.w = fmaxf(v.w, 0.f);
            }
            *(float4*)(&Alds[aLdsOff[i]]) = v;
        }

        // ---- stage B tile transposed: coalesced float4 loads, 4 b32 stores ----
#pragma unroll
        for (int i = 0; i < 4; ++i) {
            float4 v = *(const float4*)(Bk + bOff[i]);
            Blds[bLdsOff[i]              ] = v.x;
            Blds[bLdsOff[i] + 1 * BSTRIDE] = v.y;
            Blds[bLdsOff[i] + 2 * BSTRIDE] = v.z;
            Blds[bLdsOff[i] + 3 * BSTRIDE] = v.w;
        }
        Bk += (long)KC * N;

        __syncthreads();

        // ---- 16 k-steps of WMMA out of LDS ----
        const float* aFrag = &Alds[mLocal * ASTRIDE];
#pragma unroll
        for (int ks = 0; ks < KC; ks += 4) {
            const int kk = ks + kHalf;
            v2f a = *(const v2f*)(aFrag + kk);
#pragma unroll
            for (int j = 0; j < 4; ++j) {
                v2f b = *(const v2f*)(&Blds[(j * 16 + nCol) * BSTRIDE + kk]);
                acc[j] = __builtin_amdgcn_wmma_f32_16x16x4_f32(
                    false, a, false, b, (short)0, acc[j], false, false);
            }
        }
    }

    // C/D layout: VGPR r -> lanes 0-15: M=r, lanes 16-31: M=r+8; N = lane&15
    const int rBase = mBase + wave * 16 + ((lane >> 4) << 3);
#pragma unroll
    for (int j = 0; j < 4; ++j) {
        const int col = nTile + j * 16 + nCol;
        const float bv = bias ? bias[col] : 0.f;
#pragma unroll
        for (int r = 0; r < 8; ++r) {
            const int row = rBase + r;
            if (row < M) {
                float v = acc[j][r] + bv;
                if (reluOut) v = fmaxf(v, 0.f);
                C[(long)row * N + col] = v;
            }
        }
    }
}

// ---------------------------------------------------------------------------
// Degree / normalization kernels
// ---------------------------------------------------------------------------
__global__ void deg_init(float* __restrict__ deg, int Nn)
{
    int i = blockIdx.x * blockDim.x + threadIdx.x;
    if (i < Nn) deg[i] = 1.0f;                // self-loop contribution
}

__global__ void deg_edges(const long long* __restrict__ ei, float* __restrict__ deg, int E)
{
    int e = blockIdx.x * blockDim.x + threadIdx.x;
    if (e < E) atomicAdd(&deg[(int)ei[(long)E + e]], 1.0f);
}

__global__ void deg_rsqrt(float* __restrict__ deg, int Nn)
{
    int i = blockIdx.x * blockDim.x + threadIdx.x;
    if (i < Nn) deg[i] = rsqrtf(deg[i]);      // in place: deg -> dinv
}

// ---------------------------------------------------------------------------
// GCN aggregation: agg = bias + dinv^2 * hw   (self-loop fused), then
// edge scatter: agg[dst] += dinv[src]*dinv[dst] * hw[src]
// ---------------------------------------------------------------------------
__global__ void agg_init(const float* __restrict__ hw, const float* __restrict__ dinv,
                         const float* __restrict__ bias, float* __restrict__ agg,
                         int Nn, int C)
{
    long tid = (long)blockIdx.x * blockDim.x + threadIdx.x;
    const int perRow = C >> 2;
    long total = (long)Nn * perRow;
    if (tid >= total) return;
    int i  = (int)(tid / perRow);
    int cq = (int)(tid % perRow);
    float di = dinv[i];
    float nrm = di * di;
    float4 v = ((const float4*)(hw  + (long)i * C))[cq];
    float4 b = ((const float4*)bias)[cq];
    float4 o;
    o.x = b.x + nrm * v.x; o.y = b.y + nrm * v.y;
    o.z = b.z + nrm * v.z; o.w = b.w + nrm * v.w;
    ((float4*)(agg + (long)i * C))[cq] = o;
}

__global__ void scatter_edges(const long long* __restrict__ ei, const float* __restrict__ hw,
                              const float* __restrict__ dinv, float* __restrict__ agg,
                              int E, int C)
{
    long tid = (long)blockIdx.x * blockDim.x + threadIdx.x;
    const int perEdge = C >> 2;
    long total = (long)E * perEdge;
    if (tid >= total) return;
    int e  = (int)(tid / perEdge);
    int cq = (int)(tid % perEdge);
    int s = (int)ei[e];
    int d = (int)ei[(long)E + e];
    float nrm = dinv[s] * dinv[d];
    float4 v = ((const float4*)(hw + (long)s * C))[cq];
    float* out = agg + (long)d * C + (cq << 2);
    atomicAdd(out + 0, nrm * v.x);
    atomicAdd(out + 1, nrm * v.y);
    atomicAdd(out + 2, nrm * v.z);
    atomicAdd(out + 3, nrm * v.w);
}

// ---------------------------------------------------------------------------
// Global mean pool (with ReLU on the way in)
// ---------------------------------------------------------------------------
__global__ void pool_zero(float* __restrict__ sums, float* __restrict__ counts, int C)
{
    int i = blockIdx.x * blockDim.x + threadIdx.x;
    if (i < NG * C) sums[i] = 0.f;
    if (i < NG) counts[i] = 0.f;
}

__global__ void pool_count(const long long* __restrict__ batch, float* __restrict__ counts, int Nn)
{
    int i = blockIdx.x * blockDim.x + threadIdx.x;
    if (i < Nn) atomicAdd(&counts[(int)batch[i]], 1.0f);
}

__global__ void pool_accum(const float* __restrict__ agg, const long long* __restrict__ batch,
                           float* __restrict__ sums, int Nn, int C)
{
    long tid = (long)blockIdx.x * blockDim.x + threadIdx.x;
    const int perRow = C >> 2;
    long total = (long)Nn * perRow;
    if (tid >= total) return;
    int i  = (int)(tid / perRow);
    int cq = (int)(tid % perRow);
    int g  = (int)batch[i];
    float4 v = ((const float4*)(agg + (long)i * C))[cq];
    float* out = sums + (long)g * C + (cq << 2);
    atomicAdd(out + 0, fmaxf(v.x, 0.f));
    atomicAdd(out + 1, fmaxf(v.y, 0.f));
    atomicAdd(out + 2, fmaxf(v.z, 0.f));
    atomicAdd(out + 3, fmaxf(v.w, 0.f));
}

__global__ void pool_div(const float* __restrict__ sums, const float* __restrict__ counts,
                         float* __restrict__ g, int C)
{
    int i = blockIdx.x * blockDim.x + threadIdx.x;
    if (i < NG * C) {
        float cnt = fmaxf(counts[i / C], 1.0f);
        g[i] = sums[i] / cnt;
    }
}

// ---------------------------------------------------------------------------
// lin3: out[r] = g2[r,:] . W3 + b3     (G x 128 @ 128 x 1)
// ---------------------------------------------------------------------------
__global__ void lin3_kernel(const float* __restrict__ g2, const float* __restrict__ W3,
                            const float* __restrict__ b3, float* __restrict__ out, int Kd)
{
    int r = blockIdx.x * blockDim.x + threadIdx.x;
    if (r >= NG) return;
    float s = b3[0];
    const float* row = g2 + (long)r * Kd;
    for (int k = 0; k < Kd; ++k) s += row[k] * W3[k];
    out[r] = s;
}

// ---------------------------------------------------------------------------
extern "C" void kernel_launch(void* const* d_in, const int* in_sizes, int n_in,
                              void* d_out, int out_size, void* d_ws, size_t ws_size,
                              hipStream_t stream)
{
    const float*     x   = (const float*)d_in[0];
    const long long* ei  = (const long long*)d_in[1];   // [2, E] int64
    const long long* bat = (const long long*)d_in[2];   // [N] int64
    const float* W1  = (const float*)d_in[3];
    const float* b1  = (const float*)d_in[4];
    const float* Wc1 = (const float*)d_in[5];
    const float* bc1 = (const float*)d_in[6];
    const float* Wc2 = (const float*)d_in[7];
    const float* bc2 = (const float*)d_in[8];
    const float* W2  = (const float*)d_in[9];
    const float* b2  = (const float*)d_in[10];
    const float* W3  = (const float*)d_in[11];
    const float* b3  = (const float*)d_in[12];
    float* out = (float*)d_out;

    const int Nn = in_sizes[0] / IN_DIM;     // 50000
    const int E  = in_sizes[1] / 2;          // 800000

    // ---- workspace layout (fp32), h0 region reused for agg2 ----
    float* ws    = (float*)d_ws;
    float* dinv  = ws;                               // N
    float* h0    = dinv + (size_t)Nn;                // N * HID   (later: agg2)
    float* hw    = h0   + (size_t)Nn * HID;          // N * H2    (hw1 then hw2)
    float* agg1  = hw   + (size_t)Nn * H2;           // N * H2
    float* sums  = agg1 + (size_t)Nn * H2;           // NG * H2
    float* cnts  = sums + (size_t)NG * H2;           // NG
    float* gbuf  = cnts + NG;                        // NG * H2
    float* g2    = gbuf + (size_t)NG * H2;           // NG * H4
    float* agg2  = h0;                               // reuse

    const int T = 256;
    auto blocks = [](long n, int t) { return (unsigned)((n + t - 1) / t); };

    // ---- degree / symmetric normalization ----
    deg_init <<<blocks(Nn, T), T, 0, stream>>>(dinv, Nn);
    deg_edges<<<blocks(E, T),  T, 0, stream>>>(ei, dinv, E);
    deg_rsqrt<<<blocks(Nn, T), T, 0, stream>>>(dinv, Nn);

    // ---- lin1: h0 = relu(x @ W1 + b1)   [N,128]@[128,512] ----
    {
        dim3 grid((Nn + 127) / 128, HID / 64);
        gemm_wmma_f32<<<grid, 256, 0, stream>>>(x, W1, b1, h0, Nn, IN_DIM, HID, 0, 1);
    }

    // ---- conv1: hw = h0 @ Wc1 ; agg1 = bc1 + norm-scatter(hw) ----
    {
        dim3 grid((Nn + 127) / 128, H2 / 64);
        gemm_wmma_f32<<<grid, 256, 0, stream>>>(h0, Wc1, nullptr, hw, Nn, HID, H2, 0, 0);
    }
    agg_init     <<<blocks((long)Nn * (H2 / 4), T), T, 0, stream>>>(hw, dinv, bc1, agg1, Nn, H2);
    scatter_edges<<<blocks((long)E  * (H2 / 4), T), T, 0, stream>>>(ei, hw, dinv, agg1, E, H2);

    // ---- conv2: hw = relu(agg1) @ Wc2 ; agg2 = bc2 + norm-scatter(hw) ----
    {
        dim3 grid((Nn + 127) / 128, H2 / 64);
        gemm_wmma_f32<<<grid, 256, 0, stream>>>(agg1, Wc2, nullptr, hw, Nn, H2, H2, 1, 0);
    }
    agg_init     <<<blocks((long)Nn * (H2 / 4), T), T, 0, stream>>>(hw, dinv, bc2, agg2, Nn, H2);
    scatter_edges<<<blocks((long)E  * (H2 / 4), T), T, 0, stream>>>(ei, hw, dinv, agg2, E, H2);

    // ---- global mean pool over relu(agg2) ----
    pool_zero <<<blocks(NG * H2, T), T, 0, stream>>>(sums, cnts, H2);
    pool_count<<<blocks(Nn, T),      T, 0, stream>>>(bat, cnts, Nn);
    pool_accum<<<blocks((long)Nn * (H2 / 4), T), T, 0, stream>>>(agg2, bat, sums, Nn, H2);
    pool_div  <<<blocks(NG * H2, T), T, 0, stream>>>(sums, cnts, gbuf, H2);

    // ---- lin2: g2 = relu(gbuf @ W2 + b2)  [256,256]@[256,128] via WMMA ----
    {
        dim3 grid((NG + 127) / 128, H4 / 64);
        gemm_wmma_f32<<<grid, 256, 0, stream>>>(gbuf, W2, b2, g2, NG, H2, H4, 0, 1);
    }

    // ---- lin3: out = g2 @ W3 + b3 ----
    lin3_kernel<<<blocks(NG, T), T, 0, stream>>>(g2, W3, b3, out, H4);
    (void)n_in; (void)out_size; (void)ws_size;
}